// EdgeConvLayer_66632122630758
// MI455X (gfx1250) — compile-verified
//
#include <hip/hip_runtime.h>
#include <hip/hip_bf16.h>
#include <stdint.h>

// EdgeConv layer, CDNA5 / gfx1250 (wave32, WMMA, async-to-LDS).
//
//   out[b,i,o] = relu( (x@(W1-W2))[b,i,o] + bias[o] + max_{j: adj[b,i,j]!=0} (x@W2)[b,j,o] )
//
// Phase 0: Wc = [W1-W2 | W2] (128 x 256),  bias2 = [bias | 0] (256)
// Phase 1: PQ(2048x256) = X(2048x128) @ Wc + bias2   via V_WMMA_F32_16X16X4_F32
//          (cols 0..127 = p-hat, cols 128..255 = q) -- unified buffer, branchless stores
// Phase 2: masked max over j; q-tiles staged in LDS via GLOBAL_LOAD_ASYNC_TO_LDS_B128,
//          wave-uniform adj skip, relu(p-hat + m) epilogue.

#define Bb   4
#define Nn   512
#define Dd   128
#define OUTo 128
#define ROWS (Bb * Nn)      // 2048
#define PQW  256            // fused PQ row width

typedef __attribute__((ext_vector_type(2))) float v2f;
typedef __attribute__((ext_vector_type(8))) float v8f;

// ---------------- Phase 0: Wc = [W1 - W2 | W2], bias2 = [bias | 0] ----------------
__global__ __launch_bounds__(256) void edgeconv_prep(const float* __restrict__ W,
                                                     const float* __restrict__ bias,
                                                     float* __restrict__ Wc,
                                                     float* __restrict__ bias2) {
    int idx = blockIdx.x * 256 + threadIdx.x;           // over 128*128
    if (idx < Dd * OUTo) {
        int d = idx >> 7;
        int o = idx & 127;
        float w1 = W[d * OUTo + o];
        float w2 = W[(d + Dd) * OUTo + o];
        Wc[d * PQW + o]       = w1 - w2;   // p-part
        Wc[d * PQW + 128 + o] = w2;        // q-part
    }
    if (idx < PQW) bias2[idx] = (idx < OUTo) ? bias[idx] : 0.f;
}

// ---------------- Phase 1: WMMA f32 GEMM (fused, branchless epilogue) ----------------
// One 16x16 output tile per wave. 4 waves / block, 512 blocks -> 2048 tiles.
__global__ __launch_bounds__(128) void edgeconv_gemm(const float* __restrict__ X,
                                                     const float* __restrict__ Wc,
                                                     const float* __restrict__ bias2,
                                                     float* __restrict__ PQ) {
    const int wave = threadIdx.x >> 5;
    const int lane = threadIdx.x & 31;
    const int t      = blockIdx.x * 4 + wave;   // tile id, 0..2047
    const int tile_m = t >> 4;                  // 0..127  (rows of 16)
    const int tile_n = t & 15;                  // 0..15   (cols of 16 over 256)
    const int m0 = tile_m << 4;
    const int n0 = tile_n << 4;

    const int lane_lo = lane & 15;
    const int lane_hi = lane >> 4;              // 0 or 1 -> K offset {0,2}

    const int arow = m0 + lane_lo;              // A-matrix row for this lane
    const int bcol = n0 + lane_lo;              // B-matrix col for this lane
    const float* __restrict__ xrow = X + (size_t)arow * Dd;

    v8f acc = {0.f, 0.f, 0.f, 0.f, 0.f, 0.f, 0.f, 0.f};

    #pragma unroll 4
    for (int k = 0; k < Dd; k += 4) {
        const int ka = k + (lane_hi << 1);      // K base for this lane half
        // A 16x4 fragment: VGPR0=K=ka, VGPR1=K=ka+1  (float2, 8B aligned: ka even)
        v2f a = *(const v2f*)(xrow + ka);
        // B 4x16 fragment: VGPR0=K=ka, VGPR1=K=ka+1, col striped across lanes
        v2f bb;
        bb.x = Wc[ka * PQW + bcol];
        bb.y = Wc[(ka + 1) * PQW + bcol];
        acc = __builtin_amdgcn_wmma_f32_16x16x4_f32(
                /*neg_a=*/false, a, /*neg_b=*/false, bb,
                /*c_mod=*/(short)0, acc, /*reuse_a=*/false, /*reuse_b=*/false);
    }

    // D layout: VGPR r, lanes0-15 -> row m0+r, lanes16-31 -> row m0+r+8; col n0+lane_lo
    const float bv = bias2[bcol];               // zero-padded -> branchless
    float* __restrict__ dst = PQ + (size_t)(m0 + (lane_hi << 3)) * PQW + bcol;
    #pragma unroll
    for (int r = 0; r < 8; ++r)
        dst[(size_t)r * PQW] = acc[r] + bv;
}

// ---------------- Phase 2: masked max over j (async q-tile staging) ----------------
#define JT  64      // j-tile staged in LDS
#define RPB 8       // i-rows per block (= 8 waves of 256 threads)

__global__ __launch_bounds__(256) void edgeconv_reduce(const float* __restrict__ adj,
                                                       const float* __restrict__ PQ,
                                                       float* __restrict__ out) {
    __shared__ float qs[JT * OUTo];     // 64 x 128 f32 = 32 KB
    __shared__ float as[RPB * JT];      //  8 x  64 f32 =  2 KB

    const int b   = blockIdx.y;                 // batch
    const int i0  = blockIdx.x * RPB;           // row tile base
    const int tid = threadIdx.x;
    const int wave = tid >> 5;
    const int lane = tid & 31;
    const int i = i0 + wave;                    // this wave's output row

    const uint32_t qs_lds = (uint32_t)(uintptr_t)(&qs[0]);   // LDS byte offset (addr[31:0])

    const float NEG = -__builtin_inff();
    float4 m = make_float4(NEG, NEG, NEG, NEG);

    for (int j0 = 0; j0 < Nn; j0 += JT) {
        __syncthreads();                        // protect previous tile's LDS reads
        // ---- async q-tile stage: 64 rows x 128 f32 (q half of PQ, row stride 256) ----
        // 2048 float4 moves, 8 per thread, tracked on ASYNCcnt (no VGPR data round-trip)
        const float* __restrict__ qbase =
            PQ + (size_t)(b * Nn + j0) * PQW + OUTo;          // q half
        #pragma unroll
        for (int it = 0; it < 8; ++it) {
            int idx = tid + 256 * it;           // flat float4 index in tile
            int row = idx >> 5;                 // 32 float4 per tile row
            int c4  = idx & 31;
            uint64_t g = (uint64_t)(uintptr_t)(qbase + (size_t)row * PQW + c4 * 4);
            uint32_t l = qs_lds + (uint32_t)(idx << 4);
            asm volatile("global_load_async_to_lds_b128 %0, %1, off"
                         :: "v"(l), "v"(g) : "memory");
        }
        // ---- adj tile: 8 rows x 64, rows stride Nn (regular loads, tiny) ----
        const float* __restrict__ asrc =
            adj + ((size_t)(b * Nn + i0)) * Nn + j0;
        #pragma unroll
        for (int it = 0; it < 2; ++it) {
            int idx = tid + 256 * it;           // idx = r*64 + c
            int r = idx >> 6, c = idx & 63;
            as[idx] = asrc[(size_t)r * Nn + c];
        }
        asm volatile("s_wait_asynccnt 0x0" ::: "memory");
        __syncthreads();

        const float*  __restrict__ arow  = as + wave * JT;
        const float4* __restrict__ qrow4 = (const float4*)qs;
        #pragma unroll 4
        for (int jj = 0; jj < JT; ++jj) {
            // all lanes read the same LDS word -> wave-uniform branch, ~50% skipped
            if (arow[jj] != 0.f) {
                float4 qv = qrow4[jj * (OUTo / 4) + lane];
                m.x = fmaxf(m.x, qv.x);
                m.y = fmaxf(m.y, qv.y);
                m.z = fmaxf(m.z, qv.z);
                m.w = fmaxf(m.w, qv.w);
            }
        }
    }

    // out = relu(p-hat + m);  m = -inf when no neighbor -> relu gives 0
    const float4 p4 = *(const float4*)(PQ + (size_t)(b * Nn + i) * PQW + lane * 4);
    float4 o4;
    o4.x = fmaxf(p4.x + m.x, 0.f);
    o4.y = fmaxf(p4.y + m.y, 0.f);
    o4.z = fmaxf(p4.z + m.z, 0.f);
    o4.w = fmaxf(p4.w + m.w, 0.f);
    *(float4*)(out + (size_t)(b * Nn + i) * OUTo + lane * 4) = o4;
}

extern "C" void kernel_launch(void* const* d_in, const int* in_sizes, int n_in,
                              void* d_out, int out_size, void* d_ws, size_t ws_size,
                              hipStream_t stream) {
    const float* x    = (const float*)d_in[0];   // (4,512,128)
    const float* adj  = (const float*)d_in[1];   // (4,512,512)
    const float* W    = (const float*)d_in[2];   // (256,128)
    const float* bias = (const float*)d_in[3];   // (128,)
    float* out = (float*)d_out;                  // (4,512,128)

    char* ws = (char*)d_ws;
    float* Wc    = (float*)ws;                                     // 128*256 f32 = 128 KB
    float* bias2 = (float*)(ws + (size_t)Dd * PQW * 4);            // 256 f32
    float* PQ    = (float*)(ws + (size_t)Dd * PQW * 4 + 1024);     // 2048*256 f32 = 2 MB

    edgeconv_prep<<<(Dd * OUTo + 255) / 256, 256, 0, stream>>>(W, bias, Wc, bias2);
    edgeconv_gemm<<<(ROWS / 16) * (PQW / 16) / 4, 128, 0, stream>>>(x, Wc, bias2, PQ);
    edgeconv_reduce<<<dim3(Nn / RPB, Bb), 256, 0, stream>>>(adj, PQ, out);
}